// MessagePassingQuant_8022998909727
// MI455X (gfx1250) — compile-verified
//
#include <hip/hip_runtime.h>
#include <stdint.h>

#define DIM 64
#define EPSQ 1e-8f

typedef unsigned int u32;
typedef u32 __attribute__((ext_vector_type(4))) u32x4;
typedef int __attribute__((ext_vector_type(8))) i32x8;
typedef int __attribute__((ext_vector_type(4))) i32x4;

__device__ __forceinline__ float wave_max_f(float v) {
#pragma unroll
    for (int off = 16; off >= 1; off >>= 1)
        v = fmaxf(v, __shfl_xor(v, off, 32));
    return v;
}
__device__ __forceinline__ int wave_max_i(int v) {
#pragma unroll
    for (int off = 16; off >= 1; off >>= 1) {
        int o = __shfl_xor(v, off, 32);
        v = v > o ? v : o;
    }
    return v;
}

// ---------------------------------------------------------------- init
__global__ void k_init(int* __restrict__ agg, long long total, u32* __restrict__ sc) {
    long long i = (long long)blockIdx.x * blockDim.x + threadIdx.x;
    if (i < total) agg[i] = 0;
    if (i < 8) sc[i] = 0u;
}

// ---------------------------------------------------------------- per-row max|x| (TDM-staged tiles)
__global__ void k_rowmax(const float* __restrict__ x, float* __restrict__ rowmax, int N) {
    __shared__ float tile[64 * DIM];   // 16 KB; the ONLY LDS object -> offset 0
    const int row0 = blockIdx.x * 64;
    const int rows = min(64, N - row0);
    const int tid  = threadIdx.x;

#if defined(__HIP_DEVICE_COMPILE__) && __has_builtin(__builtin_amdgcn_tensor_load_to_lds) && __has_builtin(__builtin_amdgcn_s_wait_tensorcnt)
    if (tid < 32) {  // one wave issues the tensor DMA (EXEC ignored by TDM)
        unsigned long long ga = (unsigned long long)(uintptr_t)(x + (size_t)row0 * DIM);
        u32x4 g0;
        g0.x = (u32)__builtin_amdgcn_readfirstlane((int)1u);                    // count=1 (valid D#)
        g0.y = (u32)__builtin_amdgcn_readfirstlane((int)0u);                    // lds_addr = 0
        g0.z = (u32)__builtin_amdgcn_readfirstlane((int)(u32)ga);               // global_addr[31:0]
        g0.w = (u32)__builtin_amdgcn_readfirstlane((int)((u32)(ga >> 32) | (2u << 30))); // addr[56:32] | type=2
        i32x8 g1;
        g1[0] = __builtin_amdgcn_readfirstlane((int)(2u << 16));                // data_size=4B, no multicast
        g1[1] = __builtin_amdgcn_readfirstlane((int)(64u << 16));               // tensor_dim0 = 64
        g1[2] = __builtin_amdgcn_readfirstlane((int)(((u32)N & 0xffffu) << 16)); // tensor_dim1 lo
        g1[3] = __builtin_amdgcn_readfirstlane((int)((((u32)N >> 16) & 0xffffu) | (64u << 16))); // dim1 hi | tile_dim0=64
        g1[4] = __builtin_amdgcn_readfirstlane((int)(u32)rows);                 // tile_dim1 = rows (tile_dim2=0)
        g1[5] = __builtin_amdgcn_readfirstlane((int)64u);                       // tensor_dim0_stride = 64
        g1[6] = 0;
        g1[7] = 0;
        i32x4 gz4 = {0, 0, 0, 0};
        i32x8 gz8 = {0, 0, 0, 0, 0, 0, 0, 0};
        __builtin_amdgcn_tensor_load_to_lds(g0, g1, gz4, gz4, gz8, 0);
        __builtin_amdgcn_s_wait_tensorcnt(0);
    }
    __syncthreads();
#else
    for (int i = tid; i < rows * DIM; i += blockDim.x)
        tile[i] = x[(size_t)row0 * DIM + i];
    __syncthreads();
#endif

    const int r = tid >> 2, q = tid & 3;        // 4 lanes per row, 16 elems each
    if (r < rows) {
        float m = 0.f;
#pragma unroll
        for (int c = 0; c < 16; ++c)
            m = fmaxf(m, fabsf(tile[r * DIM + q * 16 + c]));
        m = fmaxf(m, __shfl_xor(m, 1, 32));
        m = fmaxf(m, __shfl_xor(m, 2, 32));
        if (q == 0) rowmax[row0 + r] = m;
    }
}

// ---------------------------------------------------------------- m1 = max over edges of rowmax[src[e]]
__global__ void k_edgemax(const long long* __restrict__ ei, const float* __restrict__ rowmax,
                          long long E, u32* __restrict__ sc) {
    long long e = (long long)blockIdx.x * blockDim.x + threadIdx.x;
    float m = 0.f;
    if (e < E) m = rowmax[(int)ei[e]];          // ei[0..E) = src
    m = wave_max_f(m);
    if ((threadIdx.x & 31) == 0) atomicMax(&sc[0], __float_as_uint(m)); // nonneg floats: uint order == float order
}

// ---------------------------------------------------------------- quantize messages + integer scatter-add
__global__ void k_scatter(const float* __restrict__ x, const long long* __restrict__ ei,
                          int* __restrict__ agg, long long E, const u32* __restrict__ sc) {
    long long t = (long long)blockIdx.x * blockDim.x + threadIdx.x;
    if (t >= E * 16) return;                    // 16 lanes per edge, 4 floats each
    long long e = t >> 4;
    int part = (int)(t & 15);
    int s = (int)ei[e];
    int d = (int)ei[E + e];
    const float* xr = x + (size_t)s * DIM + part * 4;
    __builtin_prefetch(xr, 0, 0);               // global_prefetch_b8
    float scale1 = __uint_as_float(sc[0]) / 127.0f + EPSQ;
    float inv = 1.0f / scale1;
    float4 v = *(const float4*)xr;
    int kx = (int)fmaxf(-128.f, fminf(127.f, rintf(v.x * inv)));
    int ky = (int)fmaxf(-128.f, fminf(127.f, rintf(v.y * inv)));
    int kz = (int)fmaxf(-128.f, fminf(127.f, rintf(v.z * inv)));
    int kw = (int)fmaxf(-128.f, fminf(127.f, rintf(v.w * inv)));
    int* dst = agg + (size_t)d * DIM + part * 4;
    atomicAdd(dst + 0, kx);
    atomicAdd(dst + 1, ky);
    atomicAdd(dst + 2, kz);
    atomicAdd(dst + 3, kw);
}

// ---------------------------------------------------------------- max|sum_k| over agg (integer)
__global__ void k_aggmax(const int* __restrict__ agg, long long total, u32* __restrict__ sc) {
    long long i = (long long)blockIdx.x * blockDim.x + threadIdx.x;
    int m = 0;
    if (i < total) {
        int v = agg[i];
        m = v < 0 ? -v : v;
    }
    m = wave_max_i(m);
    if ((threadIdx.x & 31) == 0) atomicMax((int*)&sc[1], m);
}

// ---------------------------------------------------------------- agg -> q2 (second fake_quant), track max|k2|
__global__ void k_quant2(int* __restrict__ buf, long long total, u32* __restrict__ sc) {
    long long i = (long long)blockIdx.x * blockDim.x + threadIdx.x;
    float scale1 = __uint_as_float(sc[0]) / 127.0f + EPSQ;
    float m2 = scale1 * (float)(int)sc[1];
    float scale2 = m2 / 127.0f + EPSQ;
    int mk = 0;
    if (i < total) {
        float agg = (float)buf[i] * scale1;
        float kf = fmaxf(-128.f, fminf(127.f, rintf(agg / scale2)));
        mk = (int)fabsf(kf);
        ((float*)buf)[i] = kf * scale2;         // q2, in place over the int accumulator
    }
    mk = wave_max_i(mk);
    if ((threadIdx.x & 31) == 0) atomicMax((int*)&sc[2], mk);
}

// ---------------------------------------------------------------- q2 -> out (third fake_quant), in place
__global__ void k_quant3(float* __restrict__ buf, long long total, const u32* __restrict__ sc) {
    long long i = (long long)blockIdx.x * blockDim.x + threadIdx.x;
    if (i >= total) return;
    float scale1 = __uint_as_float(sc[0]) / 127.0f + EPSQ;
    float m2 = scale1 * (float)(int)sc[1];
    float scale2 = m2 / 127.0f + EPSQ;
    float m3 = scale2 * (float)(int)sc[2];
    float scale3 = m3 / 127.0f + EPSQ;
    float q2 = buf[i];
    buf[i] = fmaxf(-128.f, fminf(127.f, rintf(q2 / scale3))) * scale3;
}

extern "C" void kernel_launch(void* const* d_in, const int* in_sizes, int n_in,
                              void* d_out, int out_size, void* d_ws, size_t ws_size,
                              hipStream_t stream) {
    const float* x = (const float*)d_in[0];
    const long long* ei = (const long long*)d_in[1];   // int64 [2, E]: src then dst
    const int N = in_sizes[0] / DIM;
    const long long E = (long long)in_sizes[1] / 2;
    const long long total = (long long)N * DIM;

    u32* sc = (u32*)d_ws;                               // [0]=m1 bits, [1]=max|k1|, [2]=max|k2|
    float* rowmax = (float*)((char*)d_ws + 256);
    int* agg = (int*)d_out;                             // integer accumulator aliased on d_out

    dim3 blk(256);
    unsigned gElem = (unsigned)((total + 255) / 256);

    k_init   <<<gElem, blk, 0, stream>>>(agg, total, sc);
    k_rowmax <<<(unsigned)((N + 63) / 64), blk, 0, stream>>>(x, rowmax, N);
    k_edgemax<<<(unsigned)((E + 255) / 256), blk, 0, stream>>>(ei, rowmax, E, sc);
    k_scatter<<<(unsigned)((E * 16 + 255) / 256), blk, 0, stream>>>(x, ei, agg, E, sc);
    k_aggmax <<<gElem, blk, 0, stream>>>(agg, total, sc);
    k_quant2 <<<gElem, blk, 0, stream>>>(agg, total, sc);
    k_quant3 <<<gElem, blk, 0, stream>>>((float*)d_out, total, sc);
}